// InstHead_25761213841798
// MI455X (gfx1250) — compile-verified
//
#include <hip/hip_runtime.h>

#define CDIM   32
#define BATCH  8
#define WAVES  8
#define TILEP  16
#define PTSBLK (WAVES * TILEP)   // 128 points per 256-thread block

typedef float v2f __attribute__((ext_vector_type(2)));
typedef float v8f __attribute__((ext_vector_type(8)));

// workspace layout (floats)
#define WS_HSUM  0     // [32] sum of h per channel
#define WS_HSQ   32    // [32] sum of h^2 per channel
#define WS_SCALE 64    // [32] bn scale
#define WS_SHIFT 96    // [32] bn shift
#define WS_SEG   128   // [8][32] per-batch feat sums
#define WS_CNT   384   // [8] per-batch counts
#define WS_TOTAL 392

// ---------------------------------------------------------------- pass 1
__global__ __launch_bounds__(256)
void pass1_kernel(const float* __restrict__ feats,
                  const int*   __restrict__ bidx,
                  const float* __restrict__ off_w1,
                  const float* __restrict__ off_b1,
                  const float* __restrict__ mask_w1,
                  const float* __restrict__ mask_b1,
                  const float* __restrict__ mask_w2,
                  const float* __restrict__ mask_b2,
                  float* __restrict__ ws,
                  float* __restrict__ mask_out,
                  int N)
{
    __shared__ float tile[WAVES][TILEP][CDIM + 2];   // stride 34 floats: bank-conflict free
    __shared__ float seg_s[BATCH * CDIM];
    __shared__ float cnt_s[BATCH];
    __shared__ float hsum_s[CDIM];
    __shared__ float hsq_s[CDIM];

    const int tid  = threadIdx.x;
    const int wave = tid >> 5;
    const int lane = tid & 31;
    const int nl   = lane & 15;
    const int koff = (lane < 16) ? 0 : 2;
    const int rowb = (lane < 16) ? 0 : 8;
    const int p0   = blockIdx.x * PTSBLK + wave * TILEP;

    for (int i = tid; i < BATCH * CDIM; i += 256) seg_s[i] = 0.f;
    if (tid < BATCH) cnt_s[tid] = 0.f;
    if (tid < CDIM) { hsum_s[tid] = 0.f; hsq_s[tid] = 0.f; }
    __syncthreads();

    // Phase A: coalesced channel-major staging + segment sums (batch uniform per point)
    float acc = 0.f; int curb = -1; float runlen = 0.f;
    for (int p = 0; p < TILEP; ++p) {
        int gp = p0 + p;
        float v = 0.f; int b = -1;
        if (gp < N) { v = feats[(size_t)gp * CDIM + lane]; b = bidx[gp]; }
        tile[wave][p][lane] = v;
        if (b != curb) {
            if (curb >= 0) {
                atomicAdd(&seg_s[curb * CDIM + lane], acc);
                if (lane == 0) atomicAdd(&cnt_s[curb], runlen);
            }
            curb = b; acc = 0.f; runlen = 0.f;
        }
        acc += v; runlen += 1.f;
    }
    if (curb >= 0) {
        atomicAdd(&seg_s[curb * CDIM + lane], acc);
        if (lane == 0) atomicAdd(&cnt_s[curb], runlen);
    }
    __syncthreads();

    // Phase B: h = x@W1 + b1 and m = x@Wm1 + bm1 via fp32 WMMA (K=32 -> 8 steps, 2 col tiles)
    float bo0 = off_b1[nl],  bo1 = off_b1[16 + nl];
    float bm0 = mask_b1[nl], bm1 = mask_b1[16 + nl];
    v8f h0, h1, m0, m1;
#pragma unroll
    for (int i = 0; i < 8; ++i) { h0[i] = bo0; h1[i] = bo1; m0[i] = bm0; m1[i] = bm1; }

#pragma unroll
    for (int s = 0; s < 8; ++s) {
        int k = 4 * s + koff;
        v2f a = *(const v2f*)&tile[wave][nl][4 * s + koff];
        const float* w1r = off_w1  + (size_t)k * CDIM;
        const float* wmr = mask_w1 + (size_t)k * CDIM;
        v2f b0, b1, c0, c1;
        b0[0] = w1r[nl];      b0[1] = w1r[CDIM + nl];
        b1[0] = w1r[16 + nl]; b1[1] = w1r[CDIM + 16 + nl];
        c0[0] = wmr[nl];      c0[1] = wmr[CDIM + nl];
        c1[0] = wmr[16 + nl]; c1[1] = wmr[CDIM + 16 + nl];
        h0 = __builtin_amdgcn_wmma_f32_16x16x4_f32(false, a, false, b0, (short)0, h0, false, false);
        h1 = __builtin_amdgcn_wmma_f32_16x16x4_f32(false, a, false, b1, (short)0, h1, false, false);
        m0 = __builtin_amdgcn_wmma_f32_16x16x4_f32(false, a, false, c0, (short)0, m0, false, false);
        m1 = __builtin_amdgcn_wmma_f32_16x16x4_f32(false, a, false, c1, (short)0, m1, false, false);
    }

    // BN statistics (mask out padded rows) + mask branch epilogue
    float s0 = 0.f, q0 = 0.f, s1 = 0.f, q1 = 0.f;
    float w2a = mask_w2[nl], w2b = mask_w2[16 + nl];
    float mb2 = mask_b2[0];
    float sc[8];
#pragma unroll
    for (int v = 0; v < 8; ++v) {
        bool ok = (p0 + rowb + v) < N;
        float x0 = ok ? h0[v] : 0.f;
        float x1 = ok ? h1[v] : 0.f;
        s0 += x0; q0 += x0 * x0; s1 += x1; q1 += x1 * x1;
        float r0 = fmaxf(m0[v], 0.f), r1 = fmaxf(m1[v], 0.f);
        sc[v] = r0 * w2a + r1 * w2b;
    }
    atomicAdd(&hsum_s[nl],      s0); atomicAdd(&hsq_s[nl],      q0);
    atomicAdd(&hsum_s[16 + nl], s1); atomicAdd(&hsq_s[16 + nl], q1);

#pragma unroll
    for (int v = 0; v < 8; ++v) {   // reduce over 16 channel-lanes (xor<=8 stays in half)
        sc[v] += __shfl_xor(sc[v], 1);
        sc[v] += __shfl_xor(sc[v], 2);
        sc[v] += __shfl_xor(sc[v], 4);
        sc[v] += __shfl_xor(sc[v], 8);
    }
    if (nl == 0) {
#pragma unroll
        for (int v = 0; v < 8; ++v) {
            int gp = p0 + rowb + v;
            if (gp < N) mask_out[gp] = sc[v] + mb2;
        }
    }
    __syncthreads();

    // block-level flush to global workspace
    unsafeAtomicAdd(&ws[WS_SEG + tid], seg_s[tid]);          // tid covers 8*32
    if (tid < BATCH) unsafeAtomicAdd(&ws[WS_CNT + tid], cnt_s[tid]);
    if (tid < CDIM) {
        unsafeAtomicAdd(&ws[WS_HSUM + tid], hsum_s[tid]);
        unsafeAtomicAdd(&ws[WS_HSQ  + tid], hsq_s[tid]);
    }
}

// ---------------------------------------------------------------- finalize
__global__ __launch_bounds__(256)
void finalize_kernel(float* __restrict__ ws,
                     const float* __restrict__ gamma,
                     const float* __restrict__ beta,
                     const float* __restrict__ iou_w,
                     const float* __restrict__ iou_b,
                     float* __restrict__ pooled_out,
                     float* __restrict__ iou_out,
                     float invN)
{
    int t = threadIdx.x;
    if (t < CDIM) {
        float mu  = ws[WS_HSUM + t] * invN;
        float var = fmaxf(ws[WS_HSQ + t] * invN - mu * mu, 0.f);
        float sc  = gamma[t] * rsqrtf(var + 1e-4f);
        ws[WS_SCALE + t] = sc;
        ws[WS_SHIFT + t] = beta[t] - mu * sc;
    }
    if (t < BATCH * CDIM) {
        int b = t >> 5;
        float cnt = fmaxf(ws[WS_CNT + b], 1.f);
        pooled_out[t] = ws[WS_SEG + t] / cnt;
    }
    if (t < BATCH) {
        float cnt = fmaxf(ws[WS_CNT + t], 1.f);
        float s = iou_b[0];
        for (int c = 0; c < CDIM; ++c)
            s += (ws[WS_SEG + t * CDIM + c] / cnt) * iou_w[c];
        iou_out[t] = s;
    }
}

// ---------------------------------------------------------------- pass 2
__global__ __launch_bounds__(256)
void pass2_kernel(const float* __restrict__ feats,
                  const float* __restrict__ off_w1,
                  const float* __restrict__ off_b1,
                  const float* __restrict__ off_w2,
                  const float* __restrict__ off_b2,
                  const float* __restrict__ ws,
                  float* __restrict__ off_out,   // [N][3]
                  int N)
{
    __shared__ float tile[WAVES][TILEP][CDIM + 2];

    const int tid  = threadIdx.x;
    const int wave = tid >> 5;
    const int lane = tid & 31;
    const int nl   = lane & 15;
    const int koff = (lane < 16) ? 0 : 2;
    const int rowb = (lane < 16) ? 0 : 8;
    const int p0   = blockIdx.x * PTSBLK + wave * TILEP;

    for (int p = 0; p < TILEP; ++p) {
        int gp = p0 + p;
        tile[wave][p][lane] = (gp < N) ? feats[(size_t)gp * CDIM + lane] : 0.f;
    }
    __syncthreads();

    float bo0 = off_b1[nl], bo1 = off_b1[16 + nl];
    v8f h0, h1;
#pragma unroll
    for (int i = 0; i < 8; ++i) { h0[i] = bo0; h1[i] = bo1; }

#pragma unroll
    for (int s = 0; s < 8; ++s) {
        int k = 4 * s + koff;
        v2f a = *(const v2f*)&tile[wave][nl][4 * s + koff];
        const float* w1r = off_w1 + (size_t)k * CDIM;
        v2f b0, b1;
        b0[0] = w1r[nl];      b0[1] = w1r[CDIM + nl];
        b1[0] = w1r[16 + nl]; b1[1] = w1r[CDIM + 16 + nl];
        h0 = __builtin_amdgcn_wmma_f32_16x16x4_f32(false, a, false, b0, (short)0, h0, false, false);
        h1 = __builtin_amdgcn_wmma_f32_16x16x4_f32(false, a, false, b1, (short)0, h1, false, false);
    }

    float sA = ws[WS_SCALE + nl], sB = ws[WS_SCALE + 16 + nl];
    float tA = ws[WS_SHIFT + nl], tB = ws[WS_SHIFT + 16 + nl];
    float wa0 = off_w2[nl * 3 + 0], wa1 = off_w2[nl * 3 + 1], wa2 = off_w2[nl * 3 + 2];
    float wb0 = off_w2[(16 + nl) * 3 + 0], wb1 = off_w2[(16 + nl) * 3 + 1], wb2 = off_w2[(16 + nl) * 3 + 2];

    float o0[8], o1[8], o2[8];
#pragma unroll
    for (int v = 0; v < 8; ++v) {
        float ha = fmaxf(h0[v] * sA + tA, 0.f);
        float hb = fmaxf(h1[v] * sB + tB, 0.f);
        o0[v] = ha * wa0 + hb * wb0;
        o1[v] = ha * wa1 + hb * wb1;
        o2[v] = ha * wa2 + hb * wb2;
    }
#pragma unroll
    for (int v = 0; v < 8; ++v) {
        o0[v] += __shfl_xor(o0[v], 1); o0[v] += __shfl_xor(o0[v], 2);
        o0[v] += __shfl_xor(o0[v], 4); o0[v] += __shfl_xor(o0[v], 8);
        o1[v] += __shfl_xor(o1[v], 1); o1[v] += __shfl_xor(o1[v], 2);
        o1[v] += __shfl_xor(o1[v], 4); o1[v] += __shfl_xor(o1[v], 8);
        o2[v] += __shfl_xor(o2[v], 1); o2[v] += __shfl_xor(o2[v], 2);
        o2[v] += __shfl_xor(o2[v], 4); o2[v] += __shfl_xor(o2[v], 8);
    }
    float e0 = off_b2[0], e1 = off_b2[1], e2 = off_b2[2];
    if (nl == 0) {
#pragma unroll
        for (int v = 0; v < 8; ++v) {
            int gp = p0 + rowb + v;
            if (gp < N) {
                off_out[(size_t)3 * gp + 0] = o0[v] + e0;
                off_out[(size_t)3 * gp + 1] = o1[v] + e1;
                off_out[(size_t)3 * gp + 2] = o2[v] + e2;
            }
        }
    }
}

// ---------------------------------------------------------------- launch
extern "C" void kernel_launch(void* const* d_in, const int* in_sizes, int n_in,
                              void* d_out, int out_size, void* d_ws, size_t ws_size,
                              hipStream_t stream)
{
    const float* feats   = (const float*)d_in[0];
    const int*   bidx    = (const int*)  d_in[1];
    const float* off_w1  = (const float*)d_in[2];
    const float* off_b1  = (const float*)d_in[3];
    const float* gamma   = (const float*)d_in[4];
    const float* beta    = (const float*)d_in[5];
    const float* off_w2  = (const float*)d_in[6];
    const float* off_b2  = (const float*)d_in[7];
    const float* mask_w1 = (const float*)d_in[8];
    const float* mask_b1 = (const float*)d_in[9];
    const float* mask_w2 = (const float*)d_in[10];
    const float* mask_b2 = (const float*)d_in[11];
    const float* iou_w   = (const float*)d_in[12];
    const float* iou_b   = (const float*)d_in[13];

    int N = in_sizes[0] / CDIM;
    if (N <= 0) return;

    float* out        = (float*)d_out;
    float* off_out    = out;                         // [N,3]
    float* mask_out   = out + (size_t)3 * N;         // [N,1]
    float* pooled_out = out + (size_t)4 * N;         // [8,32]
    float* iou_out    = pooled_out + BATCH * CDIM;   // [8,1]
    float* ws         = (float*)d_ws;

    hipMemsetAsync(ws, 0, WS_TOTAL * sizeof(float), stream);

    int nb = (N + PTSBLK - 1) / PTSBLK;
    pass1_kernel<<<nb, 256, 0, stream>>>(feats, bidx, off_w1, off_b1,
                                         mask_w1, mask_b1, mask_w2, mask_b2,
                                         ws, mask_out, N);
    finalize_kernel<<<1, 256, 0, stream>>>(ws, gamma, beta, iou_w, iou_b,
                                           pooled_out, iou_out, 1.0f / (float)N);
    pass2_kernel<<<nb, 256, 0, stream>>>(feats, off_w1, off_b1, off_w2, off_b2,
                                         ws, off_out, N);
}